// FraudGNN_64759516889783
// MI455X (gfx1250) — compile-verified
//
#include <hip/hip_runtime.h>

#define NT 100000
#define NA 50000
#define FDIM 256
#define HDIM 256
#define NE 800000

typedef __attribute__((ext_vector_type(2))) float v2f;
typedef __attribute__((ext_vector_type(8))) float v8f;

// ---------------------------------------------------------------------------
// Phase 1: edge scatter-add (mean aggregation numerator + counts).
// 256 threads cover the 256 features of one edge's source row; 8 edges/block.
// Coalesced gather from x_acc, global_atomic_add_f32 into the accumulator.
// ---------------------------------------------------------------------------
__global__ void fraudgnn_scatter(const float* __restrict__ x_acc,
                                 const int* __restrict__ src,
                                 const int* __restrict__ dst,
                                 float* __restrict__ summed,
                                 float* __restrict__ cnt) {
  const int tid = threadIdx.x;           // feature index 0..255
  const int e0 = blockIdx.x * 8;
#pragma unroll
  for (int i = 0; i < 8; ++i) {
    const int e = e0 + i;
    if (e < NE) {
      const int s = src[e];
      const int d = dst[e];
      const float v = x_acc[(size_t)s * FDIM + tid];
      atomicAdd(&summed[(size_t)d * FDIM + tid], v);
      if (tid == 0) atomicAdd(&cnt[d], 1.0f);
    }
  }
}

// ---------------------------------------------------------------------------
// Phase 2: tx_x = relu( (summed/cnt) @ W_l + b_l + x_tx @ W_r )
// One 512-thread block (16 waves) per 16-row stripe; A tiles staged in LDS
// once (mean scaling fused into the stage), each wave computes one 16-col
// tile via V_WMMA_F32_16X16X4_F32 reading A from LDS (ds_load_b64).
// ---------------------------------------------------------------------------
__global__ void __launch_bounds__(512)
fraudgnn_sage_wmma(const float* __restrict__ summed,
                   const float* __restrict__ cnt,
                   const float* __restrict__ x_tx,
                   const float* __restrict__ Wl,
                   const float* __restrict__ bl,
                   const float* __restrict__ Wr,
                   float* __restrict__ txx) {
  __shared__ float lds_mean[16 * FDIM];   // 16 KB
  __shared__ float lds_x[16 * FDIM];      // 16 KB
  __shared__ float inv_s[16];

  const int tid  = threadIdx.x;           // 0..511
  const int row0 = blockIdx.x * 16;

  if (tid < 16) inv_s[tid] = 1.0f / fmaxf(cnt[row0 + tid], 1.0f);
  __syncthreads();

  // Stage A tiles: 16 rows x 256 cols as float4 (1024 vec4 / 512 threads).
#pragma unroll
  for (int i = tid; i < 16 * FDIM / 4; i += 512) {
    const int r  = i >> 6;                // row within stripe
    const int c4 = (i & 63) * 4;          // col (float4 granularity)
    const float iv = inv_s[r];
    float4 m = *(const float4*)&summed[(size_t)(row0 + r) * FDIM + c4];
    float4 x = *(const float4*)&x_tx[(size_t)(row0 + r) * FDIM + c4];
    m.x *= iv; m.y *= iv; m.z *= iv; m.w *= iv;   // fused mean
    *(float4*)&lds_mean[r * FDIM + c4] = m;
    *(float4*)&lds_x[r * FDIM + c4]    = x;
  }
  __syncthreads();

  const int wave = tid >> 5;              // 0..15 -> column tile
  const int lane = tid & 31;
  const int half = lane >> 4;             // 0 or 1
  const int l15  = lane & 15;
  const int col  = wave * 16 + l15;       // B/C column for this lane

  v8f acc = {0.f, 0.f, 0.f, 0.f, 0.f, 0.f, 0.f, 0.f};

  for (int k = 0; k < FDIM; k += 4) {
    // A layout (16x4 f32): lanes 0-15 -> K={k,k+1}, lanes 16-31 -> K={k+2,k+3}
    const int kb = k + half * 2;
    const v2f a1 = *(const v2f*)&lds_mean[l15 * FDIM + kb];
    const v2f a2 = *(const v2f*)&lds_x[l15 * FDIM + kb];

    // B layout (4x16 f32): VGPR0 -> row K=kb, VGPR1 -> row K=kb+1
    v2f b1, b2;
    b1.x = Wl[(size_t)(kb + 0) * HDIM + col];
    b1.y = Wl[(size_t)(kb + 1) * HDIM + col];
    b2.x = Wr[(size_t)(kb + 0) * HDIM + col];
    b2.y = Wr[(size_t)(kb + 1) * HDIM + col];

    acc = __builtin_amdgcn_wmma_f32_16x16x4_f32(false, a1, false, b1,
                                                (short)0, acc, false, false);
    acc = __builtin_amdgcn_wmma_f32_16x16x4_f32(false, a2, false, b2,
                                                (short)0, acc, false, false);
  }

  const float bias = bl[col];
#pragma unroll
  for (int r = 0; r < 8; ++r) {
    // C/D layout: VGPR r holds rows {r, r+8}; lane group selects which.
    float v = acc[r] + bias;
    v = fmaxf(v, 0.0f);                   // relu
    const int orow = row0 + r + half * 8;
    txx[(size_t)orow * HDIM + col] = v;
  }
}

// ---------------------------------------------------------------------------
// Phase 3: logits = tx_x @ W_out + b_out. One wave per node; b128 loads,
// shuffle reduction.
// ---------------------------------------------------------------------------
__global__ void fraudgnn_logits(const float* __restrict__ txx,
                                const float* __restrict__ Wout,
                                const float* __restrict__ bout,
                                float* __restrict__ logits) {
  const int node = blockIdx.x * 8 + (threadIdx.x >> 5);
  const int lane = threadIdx.x & 31;
  float s = 0.f;
#pragma unroll
  for (int j = 0; j < 2; ++j) {
    const int f = lane * 4 + j * 128;     // lanes cover 128 contiguous floats
    const float4 x = *(const float4*)&txx[(size_t)node * HDIM + f];
    const float4 w = *(const float4*)&Wout[f];
    s += x.x * w.x + x.y * w.y + x.z * w.z + x.w * w.w;
  }
#pragma unroll
  for (int off = 16; off > 0; off >>= 1)
    s += __shfl_xor(s, off, 32);
  if (lane == 0) logits[node] = s + bout[0];
}

// ---------------------------------------------------------------------------
extern "C" void kernel_launch(void* const* d_in, const int* in_sizes, int n_in,
                              void* d_out, int out_size, void* d_ws, size_t ws_size,
                              hipStream_t stream) {
  const float* x_tx  = (const float*)d_in[0];
  const float* x_acc = (const float*)d_in[1];
  const int*   eas   = (const int*)d_in[2];   // edge_at_src (account idx)
  const int*   ead   = (const int*)d_in[3];   // edge_at_dst (tx idx)
  // d_in[4], d_in[5]: tx->acc edges (h_acc is dead code downstream; skipped)
  const float* Wl    = (const float*)d_in[6];
  const float* bl    = (const float*)d_in[7];
  const float* Wr    = (const float*)d_in[8];
  // d_in[9..11]: W_l_ta / b_l_ta / W_r_ta unused
  const float* Wout  = (const float*)d_in[12];
  const float* bout  = (const float*)d_in[13];

  float* summed = (float*)d_ws;                    // [NT, FDIM]
  float* cnt    = summed + (size_t)NT * FDIM;      // [NT]
  float* logits = (float*)d_out;                   // [NT]
  float* txx    = logits + NT;                     // [NT, HDIM]

  hipMemsetAsync(d_ws, 0, ((size_t)NT * FDIM + NT) * sizeof(float), stream);

  fraudgnn_scatter<<<NE / 8, 256, 0, stream>>>(x_acc, eas, ead, summed, cnt);

  fraudgnn_sage_wmma<<<NT / 16, 512, 0, stream>>>(summed, cnt, x_tx,
                                                  Wl, bl, Wr, txx);

  fraudgnn_logits<<<NT / 8, 256, 0, stream>>>(txx, Wout, bout, logits);
}